// DigitCaps_74294344286537
// MI455X (gfx1250) — compile-verified
//
#include <hip/hip_runtime.h>
#include <hip/hip_bf16.h>

// ---------------------------------------------------------------------------
// Capsule dynamic routing forward on gfx1250 (MI455X), fp32 WMMA path, v2.
//
//   B=256 N=1152 C=10 I=8 O=16 R=3
//
// Restructuring (b-logits are batch-invariant: update is d.mean(axis=0)):
//   s[b,(c,o)]  = sum_{(n,i)} x[b,(n,i)] * (coup[n,c]*W[n,c,o,i])  GEMM 256x160x9216
//   G[(c,o),(n,i)] = sum_b v[b,c,o]*x[b,(n,i)]                     GEMM 160x9216x256
//   blog[n,c] += (1/B) * sum_{o,i} W[n,c,o,i]*G[(c,o),(n,i)]
// u_hat (1.9 GB) never materialized; working set (~25 MB) is L2-resident.
//
// v2 changes vs v1 (same V_WMMA_F32_16X16X4_F32 core):
//  * s-GEMM K-split x16 -> 1280 waves (was 160); fixed-order reduction fused
//    with squash (deterministic, no float atomics).
//  * 2 M-tiles per wave share each Wc fragment: 3 b64 loads / 2 WMMAs.
//  * d-GEMM class loop spread across blocks: 720 blocks / 5760 waves.
//  * LDS-tiled 32x32 transpose of x (coalesced both directions).
//
// Fragment layouts per ISA (wave32):
//   A 16x4 f32 : lane = m + 16*(k>>1), reg = k&1  -> regs hold consecutive k
//   B 4x16 f32 : lane = n + 16*(k>>1), reg = k&1  -> regs hold consecutive k
//   C/D 16x16  : vgpr p holds M = p + 8*(lane>>4), N = lane&15
// ---------------------------------------------------------------------------

typedef float v2f __attribute__((ext_vector_type(2)));
typedef float v8f __attribute__((ext_vector_type(8)));

#define B_     256
#define N_     1152
#define C_     10
#define I_     8
#define O_     16
#define R_     3
#define KS_    (N_*I_)        // 9216  (reduction dim of the s-GEMM)
#define CO_    (C_*O_)        // 160
#define KSPLIT 16
#define KCHUNK (KS_/KSPLIT)   // 576 k-values = 144 WMMA steps per chunk

__device__ __forceinline__ v8f wmma_f32_k4(v2f a, v2f b, v8f c) {
  // 8 args: (neg_a, A, neg_b, B, c_mod, C, reuse_a, reuse_b)
  return __builtin_amdgcn_wmma_f32_16x16x4_f32(
      false, a, false, b, (short)0, c, false, false);
}

// ---------------------------------------------------------------- utilities
__global__ void k_zero(float* __restrict__ p, int n) {
  int i = blockIdx.x * blockDim.x + threadIdx.x;
  if (i < n) p[i] = 0.0f;
}

// LDS-tiled transpose: xT[(n,i)][b] = x[b][(n,i)]. Tiles of 32(k) x 32(b).
// grid = (KS_/32) * (B_/32) = 288*8 = 2304 blocks, 256 threads.
__global__ void k_transpose_x(const float* __restrict__ x, float* __restrict__ xT) {
  __shared__ float t[32][33];
  int kb = blockIdx.x % (KS_ / 32);     // k-tile
  int bb = blockIdx.x / (KS_ / 32);     // b-tile
  int tx = threadIdx.x & 31;            // contiguous dim
  int ty = threadIdx.x >> 5;            // 0..7
#pragma unroll
  for (int j = 0; j < 4; ++j) {
    int b = bb * 32 + ty + 8 * j;
    t[ty + 8 * j][tx] = x[(size_t)b * KS_ + kb * 32 + tx];   // coalesced read
  }
  __syncthreads();
#pragma unroll
  for (int j = 0; j < 4; ++j) {
    int k = kb * 32 + ty + 8 * j;
    xT[(size_t)k * B_ + bb * 32 + tx] = t[tx][ty + 8 * j];   // coalesced write
  }
}

// coup[n,c] = softmax over n of blog[:,c]   (one block per class)
__global__ void k_softmax(const float* __restrict__ blog, float* __restrict__ coup) {
  __shared__ float red[256];
  int c = blockIdx.x;
  int t = threadIdx.x;
  float mx = -1e30f;
  for (int n = t; n < N_; n += 256) mx = fmaxf(mx, blog[n * C_ + c]);
  red[t] = mx; __syncthreads();
  for (int s = 128; s > 0; s >>= 1) {
    if (t < s) red[t] = fmaxf(red[t], red[t + s]);
    __syncthreads();
  }
  mx = red[0]; __syncthreads();
  float sum = 0.0f;
  for (int n = t; n < N_; n += 256) sum += __expf(blog[n * C_ + c] - mx);
  red[t] = sum; __syncthreads();
  for (int s = 128; s > 0; s >>= 1) {
    if (t < s) red[t] += red[t + s];
    __syncthreads();
  }
  float inv = 1.0f / red[0];
  for (int n = t; n < N_; n += 256)
    coup[n * C_ + c] = __expf(blog[n * C_ + c] - mx) * inv;
}

// Wc[n][c][o][i] = coup[n][c] * W[n][c][o][i]   (1,474,560 elems, exact grid)
__global__ void k_scale_w(const float* __restrict__ W, const float* __restrict__ coup,
                          float* __restrict__ Wc) {
  int idx = blockIdx.x * 256 + threadIdx.x;
  int nc  = idx >> 7;                 // /(O_*I_)=128 ; nc == n*C_+c
  Wc[idx] = W[idx] * coup[nc];
}

// ------------------------------------------------------- s-GEMM (WMMA, fp32)
// Each wave: 2 M-tiles (32 batch rows) x 1 class tile x 1 K-chunk.
// Tasks = 8 m-pairs * 10 classes * 16 chunks = 1280 waves = 160 blocks.
// Per K-step: 2 A loads + 1 shared B load -> 2 WMMAs (1.5 b64 loads / WMMA).
__global__ void k_gemm_s(const float* __restrict__ x, const float* __restrict__ Wc,
                         float* __restrict__ spart) {
  int wave = threadIdx.x >> 5;
  int lane = threadIdx.x & 31;
  int id   = blockIdx.x * 8 + wave;          // 0..1279
  int mtp  = id & 7;                         // m-pair 0..7
  int c    = (id >> 3) % C_;                 // class
  int ch   = id / (8 * C_);                  // K-chunk 0..15
  int m0   = mtp * 32;
  int l15 = lane & 15, lh = lane >> 4;
  int kbeg = ch * KCHUNK, kend = kbeg + KCHUNK;

  v8f acc0 = {}, acc1 = {};
  const float* xrow0 = x + (size_t)(m0 + l15) * KS_ + 2 * lh;
  const float* xrow1 = xrow0 + (size_t)16 * KS_;
  for (int k0 = kbeg; k0 < kend; k0 += 4) {
    v2f a0 = *(const v2f*)(xrow0 + k0);                      // b64: k, k+1
    v2f a1 = *(const v2f*)(xrow1 + k0);
    int k = k0 + 2 * lh;                                     // even
    int n = k >> 3, i = k & 7;                               // i even -> i,i+1 contig
    v2f b = *(const v2f*)(Wc + (((size_t)(n * C_ + c) * O_ + l15) * I_ + i));
    acc0 = wmma_f32_k4(a0, b, acc0);
    acc1 = wmma_f32_k4(a1, b, acc1);
  }
  float* sp = spart + (size_t)ch * (B_ * CO_);
#pragma unroll
  for (int p = 0; p < 8; ++p) {
    sp[(size_t)(m0 +      p + 8 * lh) * CO_ + c * 16 + l15] = acc0[p];
    sp[(size_t)(m0 + 16 + p + 8 * lh) * CO_ + c * 16 + l15] = acc1[p];
  }
}

// Fixed-order K-split reduction fused with squash; emits v and vT[(c,o)][b].
__global__ void k_reduce_squash(const float* __restrict__ spart,
                                float* __restrict__ v, float* __restrict__ vT) {
  int idx = blockIdx.x * 256 + threadIdx.x;     // 0..40959
  float sv = 0.0f;
#pragma unroll
  for (int ch = 0; ch < KSPLIT; ++ch)
    sv += spart[(size_t)ch * (B_ * CO_) + idx];
  float sn = sv * sv + 1e-18f;
  float val = (sn / (1.0f + sn)) * sv * rsqrtf(sn);
  v[idx] = val;
  int b = idx / CO_, co = idx % CO_;
  vT[co * B_ + b] = val;
}

// ------------------------------------------- d-GEMM + contraction (fused)
// Block = (class c, 128-col strip). Each wave: one 16-col (n,i) strip
// (= 2 capsules). G_tile[o,col] = sum_b vT[(c,o),b]*xT[col,b] via WMMA
// over K=B=256, then contract with W -> blog. grid = 10*72 = 720 blocks.
__global__ void k_gemm_d(const float* __restrict__ vT, const float* __restrict__ xT,
                         const float* __restrict__ W, float* __restrict__ blog) {
  __shared__ float G[8 * 256];        // 8 waves x 16x16 tile = 8 KB
  int c  = blockIdx.x % C_;
  int cb = blockIdx.x / C_;                                   // 0..71
  int wave = threadIdx.x >> 5;
  int lane = threadIdx.x & 31;
  int l15 = lane & 15, lh = lane >> 4;
  int colbase = (cb * 8 + wave) * 16;                         // global (n,i) col
  const float* xcol = xT + (size_t)(colbase + l15) * B_ + 2 * lh;
  const float* vrow = vT + (size_t)(c * 16 + l15) * B_ + 2 * lh;

  v8f acc = {};
  for (int k0 = 0; k0 < B_; k0 += 4) {
    v2f a = *(const v2f*)(vrow + k0);
    v2f b = *(const v2f*)(xcol + k0);
    acc = wmma_f32_k4(a, b, acc);
  }
  float* g = G + wave * 256;
#pragma unroll
  for (int p = 0; p < 8; ++p)
    g[(p + 8 * lh) * 16 + l15] = acc[p];                      // rows o, cols (n,i)%16

  __syncthreads();
  // 16 capsules n per block; each 16-col wave tile holds exactly 2 of them.
  int t = threadIdx.x;
  if (t < 16) {
    int ng = cb * 16 + t;
    float d = 0.0f;
    for (int o = 0; o < O_; ++o)
#pragma unroll
      for (int i = 0; i < I_; ++i) {
        int col = t * I_ + i;                                 // 0..127
        d += W[(((size_t)(ng * C_ + c) * O_ + o) * I_) + i] *
             G[(col >> 4) * 256 + o * 16 + (col & 15)];
      }
    blog[ng * C_ + c] += d * (1.0f / B_);                     // d.mean(axis=0)
  }
}

// ---------------------------------------------------------------------------
extern "C" void kernel_launch(void* const* d_in, const int* in_sizes, int n_in,
                              void* d_out, int out_size, void* d_ws, size_t ws_size,
                              hipStream_t stream) {
  const float* x = (const float*)d_in[0];   // [B,N,I]
  const float* W = (const float*)d_in[1];   // [N,C,O,I]
  float* out = (float*)d_out;               // [B,C,O] = 40960 floats

  float* ws    = (float*)d_ws;              // ~18.4 MB total
  float* blog  = ws;                        // [N,C]          11520
  float* coup  = blog  + N_ * C_;           // [N,C]          11520
  float* Wc    = coup  + N_ * C_;           // [N,C,O,I]      1474560
  float* spart = Wc    + N_ * C_ * O_ * I_; // [KSPLIT,B,C*O] 655360
  float* vv    = spart + KSPLIT * B_ * CO_; // [B,C*O]        40960
  float* vT    = vv    + B_ * CO_;          // [C*O,B]        40960
  float* xT    = vT    + CO_ * B_;          // [N*I,B]        2359296

  k_zero<<<45, 256, 0, stream>>>(blog, N_ * C_);
  k_transpose_x<<<(KS_ / 32) * (B_ / 32), 256, 0, stream>>>(x, xT);

  for (int it = 0; it < R_; ++it) {
    k_softmax<<<C_, 256, 0, stream>>>(blog, coup);
    k_scale_w<<<(N_ * C_ * O_ * I_) / 256, 256, 0, stream>>>(W, coup, Wc);
    k_gemm_s<<<160, 256, 0, stream>>>(x, Wc, spart);
    float* vout = (it == R_ - 1) ? out : vv;
    k_reduce_squash<<<(B_ * CO_) / 256, 256, 0, stream>>>(spart, vout, vT);
    if (it < R_ - 1)
      k_gemm_d<<<C_ * (KS_ / 128), 256, 0, stream>>>(vT, xT, W, blog);
  }
}